// NodeGCN_566935683372
// MI455X (gfx1250) — compile-verified
//
#include <hip/hip_runtime.h>
#include <math.h>

typedef __attribute__((ext_vector_type(2))) float v2f;
typedef __attribute__((ext_vector_type(8))) float v8f;

#define F_IN 128
#define HID  64

// ---------------- degree / normalization ----------------

__global__ void k_deg_init(float* deg, int N) {
    int i = blockIdx.x * blockDim.x + threadIdx.x;
    if (i < N) deg[i] = 1.0f;  // self-loop contributes 1 to every node's degree
}

__global__ void k_deg_accum(const int* __restrict__ dst, float* deg, int E) {
    int e = blockIdx.x * blockDim.x + threadIdx.x;
    if (e < E) atomicAdd(&deg[dst[e]], 1.0f);
}

__global__ void k_dinv(float* deg, int N) {
    int i = blockIdx.x * blockDim.x + threadIdx.x;
    if (i < N) deg[i] = rsqrtf(deg[i]);  // deg >= 1 always (self-loops)
}

// ---------------- layer 1 GEMM: h = x @ W1 via V_WMMA_F32_16X16X4_F32 ----------------
// One wave computes one 16x16 tile of h. A: 16x4 f32 (2 VGPRs/lane),
// B: 4x16 f32 (2 VGPRs/lane), C/D: 16x16 f32 (8 VGPRs/lane).
// Lane l: M/N = l&15 ; K-pair base = (l<16) ? 0 : 2.

__global__ void k_gemm1_wmma(const float* __restrict__ x, const float* __restrict__ W1,
                             float* __restrict__ h, int N) {
    __shared__ float sW[F_IN * HID];  // 32 KB of the 320 KB WGP LDS
    // cooperative 128-bit staging of W1 into LDS
    for (int i = threadIdx.x * 4; i < F_IN * HID; i += blockDim.x * 4) {
        *(float4*)(sW + i) = *(const float4*)(W1 + i);
    }
    __syncthreads();

    const int wave = threadIdx.x >> 5;
    const int lane = threadIdx.x & 31;
    const int nColTiles = HID / 16;                       // 4
    const int nRowTiles = (N + 15) / 16;
    const int tile = blockIdx.x * 8 + wave;               // 8 waves / block
    if (tile >= nRowTiles * nColTiles) return;            // wave-uniform branch

    const int mTile = tile / nColTiles;
    const int nTile = tile % nColTiles;
    const int row0  = mTile * 16;

    const int M     = lane & 15;
    const int kbase = (lane < 16) ? 0 : 2;
    const int col   = nTile * 16 + (lane & 15);

    const int  rowA   = min(row0 + M, N - 1);             // clamp tail reads
    const float* xrow = x + (size_t)rowA * F_IN;

    v8f acc = {};
    for (int k = 0; k < F_IN; k += 4) {
        if (k + 16 < F_IN) __builtin_prefetch(xrow + k + 16, 0, 0);  // global_prefetch_b8
        v2f a, b;
        a.x = xrow[k + kbase];
        a.y = xrow[k + kbase + 1];
        b.x = sW[(k + kbase) * HID + col];
        b.y = sW[(k + kbase + 1) * HID + col];
        // D = A * B + C  (fp32 WMMA; args: neg_a, A, neg_b, B, c_mod, C, reuse_a, reuse_b)
        acc = __builtin_amdgcn_wmma_f32_16x16x4_f32(false, a, false, b,
                                                    (short)0, acc, false, false);
    }

    const int mBase = row0 + ((lane < 16) ? 0 : 8);
    #pragma unroll
    for (int v = 0; v < 8; ++v) {
        int r = mBase + v;
        if (r < N) h[(size_t)r * HID + col] = acc[v];
    }
}

// ---------------- layer 1 aggregation ----------------

// seed agg with self-loop term: agg[i] = h[i] * dinv[i]^2  (also serves as the zero-init)
__global__ void k_selfloop1(const float* __restrict__ h, const float* __restrict__ dinv,
                            float* __restrict__ agg, int N) {
    int t = blockIdx.x * blockDim.x + threadIdx.x;
    int i = t >> 4, f = (t & 15) * 4;
    if (i >= N) return;
    float w = dinv[i] * dinv[i];
    float4 hv = *(const float4*)(h + (size_t)i * HID + f);
    float4 o  = { hv.x * w, hv.y * w, hv.z * w, hv.w * w };
    *(float4*)(agg + (size_t)i * HID + f) = o;
}

// 16 lanes per edge, float4 gather + 4 global_atomic_add_f32 scatter
__global__ void k_edges1(const float* __restrict__ h, const int* __restrict__ src,
                         const int* __restrict__ dst, const float* __restrict__ dinv,
                         float* __restrict__ agg, int E) {
    int t = blockIdx.x * blockDim.x + threadIdx.x;
    int e = t >> 4, f = (t & 15) * 4;
    if (e >= E) return;
    int s = src[e], d = dst[e];
    float nrm = dinv[s] * dinv[d];
    float4 hv = *(const float4*)(h + (size_t)s * HID + f);
    float* out = agg + (size_t)d * HID + f;
    atomicAdd(out + 0, hv.x * nrm);
    atomicAdd(out + 1, hv.y * nrm);
    atomicAdd(out + 2, hv.z * nrm);
    atomicAdd(out + 3, hv.w * nrm);
}

// in-place bias + ReLU
__global__ void k_bias_relu(float* __restrict__ agg, const float* __restrict__ b1, int N) {
    int t = blockIdx.x * blockDim.x + threadIdx.x;
    int i = t >> 4, f = (t & 15) * 4;
    if (i >= N) return;
    float* p = agg + (size_t)i * HID + f;
    float4 v = *(float4*)p;
    float4 b = *(const float4*)(b1 + f);
    v.x = fmaxf(v.x + b.x, 0.0f);
    v.y = fmaxf(v.y + b.y, 0.0f);
    v.z = fmaxf(v.z + b.z, 0.0f);
    v.w = fmaxf(v.w + b.w, 0.0f);
    *(float4*)p = v;
}

// ---------------- layer 2: GEMV h2 = h1r @ W2 (64x1) ----------------

__global__ void k_gemv2(const float* __restrict__ h1, const float* __restrict__ W2,
                        float* __restrict__ h2, int N) {
    int t = blockIdx.x * blockDim.x + threadIdx.x;
    int i = t >> 4, f = (t & 15) * 4;
    if (i >= N) return;
    float4 a = *(const float4*)(h1 + (size_t)i * HID + f);
    float4 w = *(const float4*)(W2 + f);
    float p = a.x * w.x + a.y * w.y + a.z * w.z + a.w * w.w;
    // reduce across the 16-lane group (wave32)
    #pragma unroll
    for (int off = 8; off > 0; off >>= 1) p += __shfl_down(p, off, 16);
    if ((t & 15) == 0) h2[i] = p;
}

__global__ void k_selfloop2(const float* __restrict__ h2, const float* __restrict__ dinv,
                            float* __restrict__ agg2, int N) {
    int i = blockIdx.x * blockDim.x + threadIdx.x;
    if (i < N) agg2[i] = h2[i] * dinv[i] * dinv[i];
}

__global__ void k_edges2(const float* __restrict__ h2, const int* __restrict__ src,
                         const int* __restrict__ dst, const float* __restrict__ dinv,
                         float* __restrict__ agg2, int E) {
    int e = blockIdx.x * blockDim.x + threadIdx.x;
    if (e >= E) return;
    int s = src[e], d = dst[e];
    atomicAdd(&agg2[d], h2[s] * dinv[s] * dinv[d]);
}

__global__ void k_sigmoid(const float* __restrict__ agg2, const float* __restrict__ b2,
                          float* __restrict__ out, int N) {
    int i = blockIdx.x * blockDim.x + threadIdx.x;
    if (i >= N) return;
    float v = agg2[i] + b2[0];
    out[i] = 1.0f / (1.0f + expf(-v));
}

// ---------------- launcher ----------------

extern "C" void kernel_launch(void* const* d_in, const int* in_sizes, int n_in,
                              void* d_out, int out_size, void* d_ws, size_t ws_size,
                              hipStream_t stream) {
    const float* x  = (const float*)d_in[0];
    const int*   ei = (const int*)d_in[1];
    const float* W1 = (const float*)d_in[2];
    const float* b1 = (const float*)d_in[3];
    const float* W2 = (const float*)d_in[4];
    const float* b2 = (const float*)d_in[5];

    const int N = in_sizes[0] / F_IN;
    const int E = in_sizes[1] / 2;
    const int* src = ei;
    const int* dst = ei + E;

    // workspace layout (floats): dinv[N] | h[N*64] | agg1[N*64] | h2[N] | agg2[N]
    size_t Npad = (size_t)((N + 63) & ~63);
    float* dinv = (float*)d_ws;
    float* h    = dinv + Npad;
    float* agg1 = h    + (size_t)N * HID;
    float* h2   = agg1 + (size_t)N * HID;
    float* agg2 = h2   + Npad;

    const int B = 256;
    const int gN   = (N + B - 1) / B;
    const int gE   = (E + B - 1) / B;
    const int gN16 = ((size_t)N * 16 + B - 1) / B;
    const int gE16 = ((size_t)E * 16 + B - 1) / B;

    // normalization
    k_deg_init <<<gN, B, 0, stream>>>(dinv, N);
    k_deg_accum<<<gE, B, 0, stream>>>(dst, dinv, E);
    k_dinv     <<<gN, B, 0, stream>>>(dinv, N);

    // layer 1
    const int nTiles  = ((N + 15) / 16) * (HID / 16);
    const int gGemm   = (nTiles + 7) / 8;            // 8 waves per 256-thread block
    k_gemm1_wmma<<<gGemm, B, 0, stream>>>(x, W1, h, N);
    k_selfloop1 <<<gN16, B, 0, stream>>>(h, dinv, agg1, N);
    k_edges1    <<<gE16, B, 0, stream>>>(h, src, dst, dinv, agg1, E);
    k_bias_relu <<<gN16, B, 0, stream>>>(agg1, b1, N);

    // layer 2
    k_gemv2    <<<gN16, B, 0, stream>>>(agg1, W2, h2, N);
    k_selfloop2<<<gN,   B, 0, stream>>>(h2, dinv, agg2, N);
    k_edges2   <<<gE,   B, 0, stream>>>(h2, src, dst, dinv, agg2, E);
    k_sigmoid  <<<gN,   B, 0, stream>>>(agg2, b2, (float*)d_out, N);
}